// QuantizationModule_3582002724889
// MI455X (gfx1250) — compile-verified
//
#include <hip/hip_runtime.h>
#include <math.h>

// ---------------------------------------------------------------------------
// CDNA5 (gfx1250) fused JPEG quantization module.
//   - fp32 WMMA (V_WMMA_F32_16X16X4_F32): 4 independent 8x8 DCTs per 16x16
//     tile via block-diagonal basis Cbig = diag(C8, C8).
//   - Bandwidth-bound: one reduction pass + one fused transform pass.
// ---------------------------------------------------------------------------

typedef float v2f __attribute__((ext_vector_type(2)));
typedef float v8f __attribute__((ext_vector_type(8)));

#define IMG_W 672
#define IMG_H 672
#define TILES_X (IMG_W / 16)          // 42
#define TILES_Y (IMG_H / 16)          // 42
#define N_IMGS (16 * 3)               // B*C channel planes
#define N_TILES (N_IMGS * TILES_X * TILES_Y)
#define WAVES_PER_BLOCK 8
#define LDS_STRIDE 17                 // 16x16 tile, +1 pad -> conflict-free transpose
#define NPART 1024

__device__ __constant__ float QT_BASE[64] = {
    16, 11, 10, 16, 24, 40, 51, 61,
    12, 12, 14, 19, 26, 58, 60, 55,
    14, 13, 16, 24, 40, 57, 69, 56,
    14, 17, 22, 29, 51, 87, 80, 62,
    18, 22, 37, 56, 68, 109, 103, 77,
    24, 35, 55, 64, 81, 104, 113, 92,
    49, 64, 78, 87, 103, 121, 120, 101,
    72, 92, 95, 98, 112, 100, 103, 99};

// Block-diagonal 16x16 DCT basis: Cbig[r][c] = C8[r&7][c&7] if same 8x8
// quadrant, else 0.  C8[x][u] = cos((2x+1)*u*pi/16).
__device__ __forceinline__ float cbig(int r, int c) {
  if (((r ^ c) & 8) != 0) return 0.0f;
  int x = r & 7, u = c & 7;
  return __cosf((float)((2 * x + 1) * u) * 0.19634954084936207f); // pi/16
}

// ---------------------------------------------------------------------------
// Pass 1a: per-block min/max partials (float4 loads for full HBM streams)
// ---------------------------------------------------------------------------
__global__ __launch_bounds__(256) void minmax_partial_kernel(
    const float4* __restrict__ in4, long long n4, float* __restrict__ partial) {
  __shared__ float smin[256], smax[256];
  float lmin = 3.4e38f, lmax = -3.4e38f;
  for (long long i = (long long)blockIdx.x * blockDim.x + threadIdx.x; i < n4;
       i += (long long)gridDim.x * blockDim.x) {
    float4 v = in4[i];
    lmin = fminf(lmin, fminf(fminf(v.x, v.y), fminf(v.z, v.w)));
    lmax = fmaxf(lmax, fmaxf(fmaxf(v.x, v.y), fmaxf(v.z, v.w)));
  }
  int t = threadIdx.x;
  smin[t] = lmin;
  smax[t] = lmax;
  __syncthreads();
  for (int off = 128; off > 0; off >>= 1) {
    if (t < off) {
      smin[t] = fminf(smin[t], smin[t + off]);
      smax[t] = fmaxf(smax[t], smax[t + off]);
    }
    __syncthreads();
  }
  if (t == 0) {
    partial[2 * blockIdx.x + 0] = smin[0];
    partial[2 * blockIdx.x + 1] = smax[0];
  }
}

// ---------------------------------------------------------------------------
// Pass 1b: final reduction of NPART partials -> mm[0]=min, mm[1]=max
// ---------------------------------------------------------------------------
__global__ __launch_bounds__(256) void minmax_final_kernel(
    const float* __restrict__ partial, int nparts, float* __restrict__ mm) {
  __shared__ float smin[256], smax[256];
  int t = threadIdx.x;
  float lmin = 3.4e38f, lmax = -3.4e38f;
  for (int i = t; i < nparts; i += 256) {
    lmin = fminf(lmin, partial[2 * i + 0]);
    lmax = fmaxf(lmax, partial[2 * i + 1]);
  }
  smin[t] = lmin;
  smax[t] = lmax;
  __syncthreads();
  for (int off = 128; off > 0; off >>= 1) {
    if (t < off) {
      smin[t] = fminf(smin[t], smin[t + off]);
      smax[t] = fmaxf(smax[t], smax[t + off]);
    }
    __syncthreads();
  }
  if (t == 0) {
    mm[0] = smin[0];
    mm[1] = smax[0];
  }
}

// ---------------------------------------------------------------------------
// Pass 2: fused quantize -> 2D DCT -> QT round -> 2D IDCT -> dequantize.
// One wave32 owns one 16x16 pixel tile (= 4 independent 8x8 JPEG blocks).
// All four 16x16x16 matmuls are chains of 4x V_WMMA_F32_16X16X4_F32.
// Per-wave LDS scratch (16x17 f32) performs the C-layout -> A/B-layout
// transposes between stages; s_wait_dscnt 0 is the only sync needed since a
// wave's lanes are lockstep.
// ---------------------------------------------------------------------------
__global__ __launch_bounds__(256) void jpeg_tile_kernel(
    const float* __restrict__ in, float* __restrict__ out,
    const float* __restrict__ mm, int nTiles) {
  __shared__ float lds[WAVES_PER_BLOCK * 16 * LDS_STRIDE];

  const int tid = threadIdx.x;
  const int wave = tid >> 5;
  const int lane = tid & 31;
  const int half = lane >> 4;   // C/D layout: lanes 16-31 hold rows M+8
  const int n = lane & 15;      // column index owned by this lane
  float* sc = &lds[wave * 16 * LDS_STRIDE];

  const float minv = mm[0];
  const float maxv = mm[1];
  const float scalev = (maxv - minv) * (1.0f / 255.0f);
  const float invs = 255.0f / (maxv - minv);

  // Per-lane constant operand slices.
  //   WMMA k-slice s uses per-lane K indices kk = 4s + 2*half + {0,1}
  //   (A 16x4 layout: lanes 0-15 hold K,K+1; lanes 16-31 hold K+2,K+3;
  //    B 4x16 layout mirrors it).  Since the lane's A row m == its B col n:
  //   P[j] = Cbig^T[m][kk] = Cbig[kk][n]   (stage1 A, stage2 B)
  //   Q[j] = Cbig[m][kk]   = Cbig[n][kk]   (stage3 A, stage4 B)
  float P[8], Q[8], qtl[8];
#pragma unroll
  for (int s = 0; s < 4; ++s) {
#pragma unroll
    for (int v = 0; v < 2; ++v) {
      int kk = 4 * s + 2 * half + v;
      P[2 * s + v] = cbig(kk, n);
      Q[2 * s + v] = cbig(n, kk);
    }
  }
  // Scaled quantization table (quality = 10 -> scale = 500), per-lane column.
#pragma unroll
  for (int i = 0; i < 8; ++i) {
    float q = floorf((QT_BASE[i * 8 + (n & 7)] * 500.0f + 50.0f) * 0.01f);
    qtl[i] = fminf(fmaxf(q, 1.0f), 255.0f);
  }
  const float nfv = ((n & 7) == 0) ? 0.70710678118654752f : 1.0f;

  const int tileId = (int)blockIdx.x * WAVES_PER_BLOCK + wave;
  if (tileId >= nTiles) return;  // wave-uniform: EXEC stays all-ones for WMMA

  const int img = tileId / (TILES_X * TILES_Y);
  const int rem = tileId - img * (TILES_X * TILES_Y);
  const int ty = rem / TILES_X;
  const int tx = rem - ty * TILES_X;
  const size_t base =
      (size_t)img * (IMG_H * IMG_W) + (size_t)(ty * 16) * IMG_W + tx * 16 + n;
  const float* pin = in + base;
  float* pout = out + base;

  // ---- Stage 1: T1 = Cbig^T * X  (fused load + quantize + center) ----------
  v8f acc = {0.f, 0.f, 0.f, 0.f, 0.f, 0.f, 0.f, 0.f};
#pragma unroll
  for (int s = 0; s < 4; ++s) {
    const int r0 = 4 * s + 2 * half;
    float x0 = pin[(size_t)r0 * IMG_W];
    float x1 = pin[(size_t)(r0 + 1) * IMG_W];
    v2f b;
    b.x = fminf(fmaxf(rintf((x0 - minv) * invs), 0.0f), 255.0f) - 128.0f;
    b.y = fminf(fmaxf(rintf((x1 - minv) * invs), 0.0f), 255.0f) - 128.0f;
    v2f a;
    a.x = P[2 * s];
    a.y = P[2 * s + 1];
    acc = __builtin_amdgcn_wmma_f32_16x16x4_f32(false, a, false, b, (short)0,
                                                acc, false, false);
  }

  // transpose T1 via LDS (C-layout write, A-layout read)
#pragma unroll
  for (int i = 0; i < 8; ++i) sc[(i + 8 * half) * LDS_STRIDE + n] = acc[i];
  asm volatile("s_wait_dscnt 0x0" ::: "memory");

  // ---- Stage 2: D = T1 * Cbig ----------------------------------------------
  v8f acc2 = {0.f, 0.f, 0.f, 0.f, 0.f, 0.f, 0.f, 0.f};
#pragma unroll
  for (int s = 0; s < 4; ++s) {
    const int kk0 = 4 * s + 2 * half;
    v2f a;
    a.x = sc[n * LDS_STRIDE + kk0];
    a.y = sc[n * LDS_STRIDE + kk0 + 1];
    v2f b;
    b.x = P[2 * s];
    b.y = P[2 * s + 1];
    acc2 = __builtin_amdgcn_wmma_f32_16x16x4_f32(false, a, false, b, (short)0,
                                                 acc2, false, false);
  }

  // ---- Elementwise: 0.25*NF2 scale, round to QT multiples, re-apply NF2 ----
#pragma unroll
  for (int i = 0; i < 8; ++i) {
    const float nfu = (i == 0) ? 0.70710678118654752f : 1.0f;
    const float f = nfu * nfv;
    float d = 0.25f * f * acc2[i];
    d = rintf(d / qtl[i]) * qtl[i];
    acc2[i] = d * f;
  }

  // stash D' (B-layout read next: same-column redistribution via LDS)
#pragma unroll
  for (int i = 0; i < 8; ++i) sc[(i + 8 * half) * LDS_STRIDE + n] = acc2[i];
  asm volatile("s_wait_dscnt 0x0" ::: "memory");

  // ---- Stage 3: T3 = Cbig * D' ---------------------------------------------
  v8f acc3 = {0.f, 0.f, 0.f, 0.f, 0.f, 0.f, 0.f, 0.f};
#pragma unroll
  for (int s = 0; s < 4; ++s) {
    const int kk0 = 4 * s + 2 * half;
    v2f a;
    a.x = Q[2 * s];
    a.y = Q[2 * s + 1];
    v2f b;
    b.x = sc[kk0 * LDS_STRIDE + n];
    b.y = sc[(kk0 + 1) * LDS_STRIDE + n];
    acc3 = __builtin_amdgcn_wmma_f32_16x16x4_f32(false, a, false, b, (short)0,
                                                 acc3, false, false);
  }

  // transpose T3 via LDS
#pragma unroll
  for (int i = 0; i < 8; ++i) sc[(i + 8 * half) * LDS_STRIDE + n] = acc3[i];
  asm volatile("s_wait_dscnt 0x0" ::: "memory");

  // ---- Stage 4: R = T3 * Cbig^T --------------------------------------------
  v8f acc4 = {0.f, 0.f, 0.f, 0.f, 0.f, 0.f, 0.f, 0.f};
#pragma unroll
  for (int s = 0; s < 4; ++s) {
    const int kk0 = 4 * s + 2 * half;
    v2f a;
    a.x = sc[n * LDS_STRIDE + kk0];
    a.y = sc[n * LDS_STRIDE + kk0 + 1];
    v2f b;
    b.x = Q[2 * s];
    b.y = Q[2 * s + 1];
    acc4 = __builtin_amdgcn_wmma_f32_16x16x4_f32(false, a, false, b, (short)0,
                                                 acc4, false, false);
  }

  // ---- Fused un-center + dequantize + store --------------------------------
#pragma unroll
  for (int i = 0; i < 8; ++i) {
    float r = (0.25f * acc4[i] + 128.0f) * scalev + minv;
    pout[(size_t)(i + 8 * half) * IMG_W] = r;
  }
}

// ---------------------------------------------------------------------------
extern "C" void kernel_launch(void* const* d_in, const int* in_sizes, int n_in,
                              void* d_out, int out_size, void* d_ws,
                              size_t ws_size, hipStream_t stream) {
  const float* in = (const float*)d_in[0];
  float* out = (float*)d_out;
  const long long n = (long long)in_sizes[0];  // 16*3*672*672

  float* mm = (float*)d_ws;       // [0]=min, [1]=max
  float* partial = mm + 2;        // NPART float2 partials

  minmax_partial_kernel<<<NPART, 256, 0, stream>>>((const float4*)in, n / 4,
                                                   partial);
  minmax_final_kernel<<<1, 256, 0, stream>>>(partial, NPART, mm);

  const int nTiles = N_TILES;
  const int blocks = (nTiles + WAVES_PER_BLOCK - 1) / WAVES_PER_BLOCK;
  jpeg_tile_kernel<<<blocks, 256, 0, stream>>>(in, out, mm, nTiles);
}